// GNNLayer_53412213293592
// MI455X (gfx1250) — compile-verified
//
#include <hip/hip_runtime.h>

#define D 256  // D_IN == D_OUT == 256

typedef __attribute__((ext_vector_type(2))) float v2f;
typedef __attribute__((ext_vector_type(8))) float v8f;
typedef __attribute__((ext_vector_type(4))) unsigned int v4u;
typedef __attribute__((ext_vector_type(4))) int v4i;
typedef __attribute__((ext_vector_type(8))) int v8i;

// ---------------------------------------------------------------------------
// TDM: DMA a 2-D tile (tileW x tileH fp32, row-major, rowStride elements)
// from global memory into LDS at ldsByteAddr. Reads past tensorW/tensorHrem
// (dims measured from tile start) return zero -> free OOB handling.
// D# layout per CDNA5 ISA ch.8 (group0: count/lds/global/type, group1: dims).
// 6-arg builtin variant (clang-23 / therock headers).
// ---------------------------------------------------------------------------
__device__ __forceinline__ void tdm_load_tile_f32(
    const float* gptr, unsigned ldsByteAddr,
    unsigned tensorW, unsigned tensorHrem, unsigned rowStride,
    unsigned tileW, unsigned tileH)
{
    unsigned long long ga = (unsigned long long)(size_t)gptr;
    v4u g0;
    g0[0] = 1u;                                   // count=1, is_restore=0
    g0[1] = ldsByteAddr;                          // lds_addr
    g0[2] = (unsigned)ga;                         // global_addr[31:0]
    g0[3] = (unsigned)(ga >> 32) | (2u << 30);    // global_addr[56:32] | type=2
    v8i g1;
    g1[0] = (int)(2u << 16);                      // wg_mask=0, data_size=2 (4B)
    g1[1] = (int)((tensorW & 0xFFFFu) << 16);     // tensor_dim0[15:0]
    g1[2] = (int)((tensorW >> 16) | ((tensorHrem & 0xFFFFu) << 16)); // td0 hi | td1 lo
    g1[3] = (int)((tensorHrem >> 16) | (tileW << 16));               // td1 hi | tile_dim0
    g1[4] = (int)tileH;                           // tile_dim1, tile_dim2=0
    g1[5] = (int)rowStride;                       // tensor_dim0_stride[31:0]
    g1[6] = 0;                                    // stride0 hi | stride1 lo
    g1[7] = 0;                                    // stride1 hi
    v4i z4 = {0, 0, 0, 0};
    v8i z8 = {0, 0, 0, 0, 0, 0, 0, 0};
    __builtin_amdgcn_tensor_load_to_lds(g0, g1, z4, z4, z8, 0);
}

// ---------------------------------------------------------------------------
// Kernel 1: support = features @ weight  (fp32 WMMA 16x16x4, TDM-staged LDS)
// block = 256 threads = 8 waves; block tile = 128 rows x 64 cols
// per-wave tile = 16 rows x 64 cols (4 accumulators of v8f)
// K chunked by 64, double-buffered: ldsA 2x(128x64), ldsW 2x(64x64) = 96 KB
// ---------------------------------------------------------------------------
__global__ __launch_bounds__(256) void gnn_gemm_wmma_f32(
    const float* __restrict__ A,   // [M][256] features
    const float* __restrict__ W,   // [256][256] weight
    float* __restrict__ S,         // [M][256] support (workspace)
    int M)
{
    __shared__ float smem[2 * (128 * 64) + 2 * (64 * 64)];   // 96 KB
    float* const bufA0 = smem;
    float* const bufA1 = smem + 8192;
    float* const bufW0 = smem + 16384;
    float* const bufW1 = smem + 16384 + 4096;
    const unsigned smemBase = (unsigned)(size_t)smem;        // LDS byte offset
    const unsigned addrA0 = smemBase;
    const unsigned addrA1 = smemBase + 8192u * 4u;
    const unsigned addrW0 = smemBase + 16384u * 4u;
    const unsigned addrW1 = smemBase + (16384u + 4096u) * 4u;

    const int wave = threadIdx.x >> 5;
    const int lane = threadIdx.x & 31;
    const int half = lane >> 4;       // 0: lanes 0-15, 1: lanes 16-31
    const int lm   = lane & 15;

    const int rowBase = blockIdx.x * 128;
    const int colBase = blockIdx.y * 64;
    const unsigned rowsRem = (unsigned)(M - rowBase);        // OOB rows -> 0

    const bool issuer = (wave == 0);  // wave-uniform guard: TDM ignores EXEC,
                                      // so non-issuing waves must branch PAST it

    v8f acc0 = {}, acc1 = {}, acc2 = {}, acc3 = {};

    // prologue: DMA chunk 0 (A tile 64x128, W tile 64x64)
    if (issuer) {
        tdm_load_tile_f32(A + (size_t)rowBase * D, addrA0,
                          D, rowsRem, D, 64, 128);
        tdm_load_tile_f32(W + colBase, addrW0,
                          (unsigned)(D - colBase), D, D, 64, 64);
    }

    for (int i = 0; i < 4; ++i) {                 // 4 K-chunks of 64
        const int cur = i & 1;
        // issue next chunk into the other buffer before consuming this one
        if (i < 3 && issuer) {
            const int kc2 = (i + 1) * 64;
            tdm_load_tile_f32(A + (size_t)rowBase * D + kc2,
                              cur ? addrA0 : addrA1,
                              (unsigned)(D - kc2), rowsRem, D, 64, 128);
            tdm_load_tile_f32(W + (size_t)kc2 * D + colBase,
                              cur ? addrW0 : addrW1,
                              (unsigned)(D - colBase), (unsigned)(D - kc2),
                              D, 64, 64);
        }
        if (issuer) {
            // tensor ops from one wave complete in order: waiting for <=2
            // outstanding guarantees chunk i's two DMAs have landed
            if (i < 3) __builtin_amdgcn_s_wait_tensorcnt(2);
            else       __builtin_amdgcn_s_wait_tensorcnt(0);
        }
        __syncthreads();                          // publish LDS tile to all waves

        const float* ldsA = cur ? bufA1 : bufA0;  // [128][64]
        const float* ldsW = cur ? bufW1 : bufW0;  // [64][64]
        const int mL = wave * 16 + lm;            // local row this lane owns
        #pragma unroll
        for (int kk = 0; kk < 64; kk += 4) {
            // A fragment 16x4: lanes 0-15 -> K=kk,kk+1 ; lanes 16-31 -> kk+2,kk+3
            const int ka = kk + half * 2;
            v2f a;
            a.x = ldsA[mL * 64 + ka];
            a.y = ldsA[mL * 64 + ka + 1];
            // B fragments 4x16 per 16-col tile (VGPR0: K=ka, VGPR1: K=ka+1)
            v2f b0, b1, b2, b3;
            b0.x = ldsW[ka * 64 +  0 + lm];  b0.y = ldsW[(ka + 1) * 64 +  0 + lm];
            b1.x = ldsW[ka * 64 + 16 + lm];  b1.y = ldsW[(ka + 1) * 64 + 16 + lm];
            b2.x = ldsW[ka * 64 + 32 + lm];  b2.y = ldsW[(ka + 1) * 64 + 32 + lm];
            b3.x = ldsW[ka * 64 + 48 + lm];  b3.y = ldsW[(ka + 1) * 64 + 48 + lm];

            acc0 = __builtin_amdgcn_wmma_f32_16x16x4_f32(false, a, false, b0, (short)0, acc0, false, false);
            acc1 = __builtin_amdgcn_wmma_f32_16x16x4_f32(false, a, false, b1, (short)0, acc1, false, false);
            acc2 = __builtin_amdgcn_wmma_f32_16x16x4_f32(false, a, false, b2, (short)0, acc2, false, false);
            acc3 = __builtin_amdgcn_wmma_f32_16x16x4_f32(false, a, false, b3, (short)0, acc3, false, false);
        }
        __syncthreads();                          // buffer reuse fence
    }

    // C/D layout: VGPR v, lanes 0-15 -> M=v, lanes 16-31 -> M=v+8, N=lane%16
    const int rTop = rowBase + wave * 16 + half * 8;
    const int c0   = colBase + lm;
    #pragma unroll
    for (int v = 0; v < 8; ++v) {
        int gr = rTop + v;
        if (gr < M) {
            S[(size_t)gr * D + c0     ] = acc0[v];
            S[(size_t)gr * D + c0 + 16] = acc1[v];
            S[(size_t)gr * D + c0 + 32] = acc2[v];
            S[(size_t)gr * D + c0 + 48] = acc3[v];
        }
    }
}

// ---------------------------------------------------------------------------
// Kernel 2: edge-parallel scatter: out[row[e]] += val[e] * support[col[e]]
// 64 threads per edge, float4 gather + 4x global_atomic_add_f32.
// support (102.4 MB) and out (102.4 MB) are ~L2-resident (192 MB L2);
// atomics resolve at the L2 atomic units.
// ---------------------------------------------------------------------------
__global__ __launch_bounds__(256) void gnn_spmm_scatter(
    const float* __restrict__ S,
    const long long* __restrict__ rows,
    const long long* __restrict__ cols,
    const float* __restrict__ vals,
    float* __restrict__ out,
    long long E)
{
    const long long tid      = (long long)blockIdx.x * blockDim.x + threadIdx.x;
    const long long nthreads = (long long)gridDim.x * blockDim.x;
    const int t = (int)(tid & 63);           // 64 threads per edge
    long long e = tid >> 6;
    const long long estep = nthreads >> 6;

    for (; e < E; e += estep) {
        const long long c = cols[e];
        const long long r = rows[e];
        const float v = vals[e];
        float4 g = *(const float4*)(S + c * D + t * 4);
        float* o = out + r * D + t * 4;
        atomicAdd(o + 0, v * g.x);
        atomicAdd(o + 1, v * g.y);
        atomicAdd(o + 2, v * g.z);
        atomicAdd(o + 3, v * g.w);
    }
}

// ---------------------------------------------------------------------------
// Kernel 3: in-place ReLU (vectorized float4)
// ---------------------------------------------------------------------------
__global__ __launch_bounds__(256) void gnn_relu_inplace(float* __restrict__ out,
                                                        long long n4)
{
    long long i = (long long)blockIdx.x * blockDim.x + threadIdx.x;
    const long long s = (long long)gridDim.x * blockDim.x;
    for (; i < n4; i += s) {
        float4 v = ((float4*)out)[i];
        v.x = fmaxf(v.x, 0.f);
        v.y = fmaxf(v.y, 0.f);
        v.z = fmaxf(v.z, 0.f);
        v.w = fmaxf(v.w, 0.f);
        ((float4*)out)[i] = v;
    }
}

extern "C" void kernel_launch(void* const* d_in, const int* in_sizes, int n_in,
                              void* d_out, int out_size, void* d_ws, size_t ws_size,
                              hipStream_t stream) {
    const float*     features = (const float*)d_in[0];      // [M][256] fp32
    const float*     weight   = (const float*)d_in[1];      // [256][256] fp32
    const long long* adj_row  = (const long long*)d_in[2];  // [E] int64
    const long long* adj_col  = (const long long*)d_in[3];  // [E] int64
    const float*     adj_val  = (const float*)d_in[4];      // [E] fp32

    const int       M = in_sizes[0] / D;      // 100000 nodes
    const long long E = in_sizes[2];          // 3.2M edges

    float* out     = (float*)d_out;
    float* support = (float*)d_ws;            // needs M*D*4 = 102.4 MB scratch

    // zero the accumulator (memset node is graph-capturable)
    (void)hipMemsetAsync(d_out, 0, (size_t)M * D * sizeof(float), stream);

    // 1) support = X @ W  (WMMA fp32, TDM-staged tiles)
    dim3 gg((M + 127) / 128, D / 64);
    gnn_gemm_wmma_f32<<<gg, 256, 0, stream>>>(features, weight, support, M);

    // 2) scatter-accumulate over edges
    gnn_spmm_scatter<<<8192, 256, 0, stream>>>(support, adj_row, adj_col,
                                               adj_val, out, E);

    // 3) ReLU in place
    const long long n4 = (long long)M * D / 4;
    gnn_relu_inplace<<<4096, 256, 0, stream>>>(out, n4);
}